// ResNetWeldonEmbedding_8074538516818
// MI455X (gfx1250) — compile-verified
//
#include <hip/hip_runtime.h>
#include <math.h>

// ---------------- problem constants ----------------
#define IMGS   16
#define EMB    2400
#define HW     196          // 14*14
#define NCOL   (IMGS*HW)    // 3136
#define CAPS   64
#define ATT_K  180
#define NPOOL  15

typedef __attribute__((ext_vector_type(2))) float v2f;
typedef __attribute__((ext_vector_type(8))) float v8f;

enum { EPI_NONE = 0, EPI_ABS = 1, EPI_BIAS = 2, EPI_SCALE_ROW = 3 };

__device__ __forceinline__ v2f ld2(const float* p) { return *(const v2f*)p; }

__device__ __forceinline__ v8f wmma4(v2f a, v2f b, v8f c) {
    return __builtin_amdgcn_wmma_f32_16x16x4_f32(false, a, false, b,
                                                 (short)0, c, false, false);
}

// ---------------------------------------------------------------
// f32 WMMA GEMM, both operands K-contiguous:
//   C[m,n] = epi( sum_k A[m*As0 + k] * B[n*Bs1 + k] )
// One wave computes a 32x32 tile (4 accumulators, A/B fragments reused 2x);
// every fragment is a single global_load_b64 (K pairs are 8B-aligned).
// Block = 128 threads = 4 waves -> 4 consecutive 32-wide n-tiles.
// Requires N%32==0, K%4==0, As0/Bs1 even; M=16 handled via m2 store guard.
// ---------------------------------------------------------------
template <int EPI>
__global__ __launch_bounds__(128)
void gemm_f32_wmma(const float* __restrict__ A, int As0, int Abatch,
                   const float* __restrict__ B, int Bs1, int Bbatch,
                   float*       __restrict__ C, int Cs0, int Cbatch,
                   const float* __restrict__ extra, int ExtraBatch,
                   int M, int N, int K)
{
    const int z = blockIdx.z;
    A += (size_t)z * (size_t)Abatch;
    B += (size_t)z * (size_t)Bbatch;
    C += (size_t)z * (size_t)Cbatch;
    if (EPI == EPI_BIAS || EPI == EPI_SCALE_ROW)
        extra += (size_t)z * (size_t)ExtraBatch;

    const int wave = threadIdx.x >> 5;
    const int lane = threadIdx.x & 31;
    const int l16  = lane & 15;
    const int koff = (lane >> 4) << 1;   // 0 for lanes 0-15, 2 for lanes 16-31

    const int n0 = (blockIdx.x * 4 + wave) * 32;
    if (n0 >= N) return;                 // wave-uniform: EXEC stays all-ones
    const int m0 = blockIdx.y * 32;
    const bool m2 = (m0 + 16) < M;       // second m-half valid (wave-uniform)

    const float* arow0 = A + (size_t)(m0 + l16) * (size_t)As0;
    // if second half invalid, alias half 0 (loads stay in-bounds; stores guarded)
    const float* arow1 = m2 ? arow0 + (size_t)16 * (size_t)As0 : arow0;
    const float* bp0   = B + (size_t)(n0 + l16) * (size_t)Bs1;
    const float* bp1   = bp0 + (size_t)16 * (size_t)Bs1;

    v8f acc00 = {0.f,0.f,0.f,0.f,0.f,0.f,0.f,0.f};
    v8f acc01 = acc00, acc10 = acc00, acc11 = acc00;

    for (int k = 0; k < K; k += 4) {
        const int kk = k + koff;
        v2f a0 = ld2(arow0 + kk);
        v2f a1 = ld2(arow1 + kk);
        v2f b0 = ld2(bp0 + kk);
        v2f b1 = ld2(bp1 + kk);
        acc00 = wmma4(a0, b0, acc00);
        acc01 = wmma4(a0, b1, acc01);
        acc10 = wmma4(a1, b0, acc10);
        acc11 = wmma4(a1, b1, acc11);
    }

    // C/D layout: VGPR r -> row tileM + r + (lane>=16 ? 8 : 0), col tileN + l16
    const int half8 = (lane >> 4) << 3;
#pragma unroll
    for (int r = 0; r < 8; ++r) {
        const int mA = m0 + r + half8;        // m-half 0
        const int mB = mA + 16;               // m-half 1
        const int nA = n0 + l16;              // n-half 0
        const int nB = nA + 16;               // n-half 1
        float v00 = acc00[r], v01 = acc01[r], v10 = acc10[r], v11 = acc11[r];
        if (EPI == EPI_ABS) {
            v00 = fabsf(v00); v01 = fabsf(v01); v10 = fabsf(v10); v11 = fabsf(v11);
        } else if (EPI == EPI_BIAS) {
            v00 += extra[nA]; v01 += extra[nB]; v10 += extra[nA]; v11 += extra[nB];
        } else if (EPI == EPI_SCALE_ROW) {
            v00 *= extra[mA]; v01 *= extra[mA]; v10 *= extra[mB]; v11 *= extra[mB];
        }
        C[(size_t)mA * (size_t)Cs0 + (size_t)nA] = v00;
        C[(size_t)mA * (size_t)Cs0 + (size_t)nB] = v01;
        if (m2) {
            C[(size_t)mB * (size_t)Cs0 + (size_t)nA] = v10;
            C[(size_t)mB * (size_t)Cs0 + (size_t)nB] = v11;
        }
    }
}

// ---------------------------------------------------------------
// Gtt[(i*196+p)*2400 + e] = g[(i*EMB + e)*196 + p]
// (K(=e)-major layout so the dominant GEMM gets b64 K-pair loads)
// ---------------------------------------------------------------
__global__ __launch_bounds__(256)
void transpose_kernel(const float* __restrict__ g, float* __restrict__ Gtt)
{
    int o = blockIdx.x * 256 + threadIdx.x;
    const int total = NCOL * EMB;
    if (o >= total) return;
    int col = o / EMB;              // i*196 + p
    int e   = o - col * EMB;
    int i   = col / HW;
    int p   = col - i * HW;
    Gtt[o] = g[((size_t)i * EMB + (size_t)e) * HW + (size_t)p];
}

// ---------------------------------------------------------------
// Weldon pooling: mean(top-15) + mean(bottom-15) of 196 spatial values
// One thread per (img, channel); insertion arrays fully unrolled -> registers.
// ---------------------------------------------------------------
__global__ __launch_bounds__(256)
void weldon_kernel(const float* __restrict__ g, float* __restrict__ hpool)
{
    int idx = blockIdx.x * 256 + threadIdx.x;
    if (idx >= IMGS * EMB) return;
    const float* p = g + (size_t)idx * HW;

    float top[NPOOL], bot[NPOOL];
#pragma unroll
    for (int j = 0; j < NPOOL; ++j) { top[j] = -INFINITY; bot[j] = INFINITY; }

    for (int s = 0; s < HW; ++s) {
        float v = p[s];
        if (v > top[0]) {           // top kept ascending, min at [0]
            top[0] = v;
#pragma unroll
            for (int j = 0; j < NPOOL - 1; ++j)
                if (top[j] > top[j + 1]) { float t = top[j]; top[j] = top[j + 1]; top[j + 1] = t; }
        }
        if (v < bot[0]) {           // bottom kept descending, max at [0]
            bot[0] = v;
#pragma unroll
            for (int j = 0; j < NPOOL - 1; ++j)
                if (bot[j] < bot[j + 1]) { float t = bot[j]; bot[j] = bot[j + 1]; bot[j + 1] = t; }
        }
    }
    float st = 0.f, sb = 0.f;
#pragma unroll
    for (int j = 0; j < NPOOL; ++j) { st += top[j]; sb += bot[j]; }
    hpool[idx] = (st + sb) * (1.0f / (float)NPOOL);
}

// ---------------------------------------------------------------
// Exact top-180 selection per caption via LDS radix-select on
// order-preserving uint keys. W[c,e] = |caps[c,e]| iff e is in the
// top-180 of row c (stable: ties take lowest index first, like argsort).
// ---------------------------------------------------------------
__global__ __launch_bounds__(256)
void topk_kernel(const float* __restrict__ caps, float* __restrict__ W)
{
    const int c = blockIdx.x;
    const int tid = threadIdx.x;
    __shared__ unsigned keys[EMB];
    __shared__ unsigned hist[256];
    __shared__ unsigned s_sel, s_want;

    const float* row = caps + (size_t)c * EMB;
    for (int e = tid; e < EMB; e += 256) {
        unsigned b = __float_as_uint(row[e]);
        keys[e] = (b & 0x80000000u) ? ~b : (b | 0x80000000u);  // larger float -> larger key
    }
    if (tid == 0) s_want = ATT_K;
    __syncthreads();

    unsigned prefmask = 0u, prefval = 0u;
    for (int shift = 24; shift >= 0; shift -= 8) {
        hist[tid] = 0u;
        __syncthreads();
        for (int e = tid; e < EMB; e += 256) {
            unsigned k = keys[e];
            if ((k & prefmask) == prefval) atomicAdd(&hist[(k >> shift) & 255u], 1u);
        }
        __syncthreads();
        if (tid == 0) {
            unsigned w = s_want, cum = 0u, sel = 0u;
            for (int d = 255; d >= 0; --d) {
                unsigned h = hist[d];
                if (cum + h >= w) { sel = (unsigned)d; s_want = w - cum; break; }
                cum += h;
            }
            s_sel = sel;
        }
        __syncthreads();
        prefval  |= (s_sel << shift);
        prefmask |= (255u << shift);
        __syncthreads();
    }

    const unsigned T = prefval;   // key of the 180th largest element
    for (int e = tid; e < EMB; e += 256)
        W[(size_t)c * EMB + e] = (keys[e] > T) ? fabsf(row[e]) : 0.0f;
    __syncthreads();
    if (tid == 0) {               // include first `s_want` elements equal to T
        unsigned rem = s_want;
        for (int e = 0; e < EMB && rem; ++e)
            if (keys[e] == T) { W[(size_t)c * EMB + e] = fabsf(row[e]); --rem; }
    }
}

// ---------------------------------------------------------------
// Sinv[i*64+c] = 1 / sum_p heatU[c*3136 + i*196 + p]
// ---------------------------------------------------------------
__global__ __launch_bounds__(64)
void sums_kernel(const float* __restrict__ heatU, float* __restrict__ Sinv)
{
    int b = blockIdx.x;            // i*64 + c
    int i = b >> 6, c = b & 63;
    int tid = threadIdx.x;
    const float* p = heatU + (size_t)c * NCOL + (size_t)i * HW;
    float s = 0.f;
    for (int e = tid; e < HW; e += 64) s += p[e];
    __shared__ float red[64];
    red[tid] = s;
    __syncthreads();
    for (int off = 32; off > 0; off >>= 1) {
        if (tid < off) red[tid] += red[tid + off];
        __syncthreads();
    }
    if (tid == 0) Sinv[b] = 1.0f / red[0];
}

// ---------------------------------------------------------------
extern "C" void kernel_launch(void* const* d_in, const int* in_sizes, int n_in,
                              void* d_out, int out_size, void* d_ws, size_t ws_size,
                              hipStream_t stream)
{
    const float* g    = (const float*)d_in[0];   // [16, 2400, 14, 14]
    const float* caps = (const float*)d_in[1];   // [64, 2400]
    const float* fc_w = (const float*)d_in[2];   // [2400, 2400]
    const float* fc_b = (const float*)d_in[3];   // [2400]
    float* out = (float*)d_out;

    // output layout: x [16*2400] | xa [16*64*2400] | g [16*2400*196]
    float* out_x  = out;
    float* out_xa = out + (size_t)IMGS * EMB;
    float* out_g  = out_xa + (size_t)IMGS * CAPS * EMB;

    // workspace layout (floats)
    float* ws    = (float*)d_ws;
    float* Gtt   = ws;                                 // [3136, 2400]  g, e-contiguous
    float* agfT  = Gtt   + (size_t)NCOL * EMB;         // [3136, 2400]  |fc_w @ g|^T
    float* hpool = agfT  + (size_t)NCOL * EMB;         // [16, 2400]
    float* W     = hpool + (size_t)IMGS * EMB;         // [64, 2400]
    float* heatU = W     + (size_t)CAPS * EMB;         // [64, 3136]
    float* Sinv  = heatU + (size_t)CAPS * NCOL;        // [1024]
    float* xa1   = Sinv  + (size_t)IMGS * CAPS;        // [16, 64, 2400]

    // 1) transpose g -> Gtt[col, e]
    {
        int total = NCOL * EMB;
        transpose_kernel<<<(total + 255) / 256, 256, 0, stream>>>(g, Gtt);
    }
    // 2) Weldon pooling -> hpool
    {
        int total = IMGS * EMB;
        weldon_kernel<<<(total + 255) / 256, 256, 0, stream>>>(g, hpool);
    }
    // 3) top-180 per caption -> W
    topk_kernel<<<CAPS, 256, 0, stream>>>(caps, W);

    // 4) x = hpool @ fc_w^T + fc_b   (M=16, N=2400, K=2400)
    {
        dim3 grid((75 + 3) / 4, 1, 1);          // 2400/32 = 75 n-tiles
        gemm_f32_wmma<EPI_BIAS><<<grid, 128, 0, stream>>>(
            hpool, EMB, 0,
            fc_w,  EMB, 0,        // B[k=e, n=o] = fc_w[o*EMB + e]
            out_x, EMB, 0,
            fc_b, 0,
            IMGS, EMB, EMB);
    }
    // 5) agfT[col, o] = |sum_e Gtt[col,e] * fc_w[o,e]|   (M=3136, N=2400, K=2400)
    //    dominant GEMM; A and B both K-contiguous, coalesced row-major store
    {
        dim3 grid((75 + 3) / 4, NCOL / 32, 1);   // 75 n-tiles, 98 m-tiles
        gemm_f32_wmma<EPI_ABS><<<grid, 128, 0, stream>>>(
            Gtt,  EMB, 0,
            fc_w, EMB, 0,
            agfT, EMB, 0,
            nullptr, 0,
            NCOL, EMB, EMB);
    }
    // 6) heatU[c, col] = sum_e W[c,e] * agfT[col,e]   (M=64, N=3136, K=2400)
    {
        dim3 grid(((NCOL / 32) + 3) / 4, CAPS / 32, 1);
        gemm_f32_wmma<EPI_NONE><<<grid, 128, 0, stream>>>(
            W,    EMB, 0,
            agfT, EMB, 0,
            heatU, NCOL, 0,
            nullptr, 0,
            CAPS, NCOL, EMB);
    }
    // 7) per-(img,cap) spatial sums -> 1/S
    sums_kernel<<<IMGS * CAPS, 64, 0, stream>>>(heatU, Sinv);

    // 8) xa1[i,c,e] = (1/S[i,c]) * sum_p heatU[c, i*196+p] * g[i,e,p]
    //    (batched, M=64, N=2400, K=196; B is the ORIGINAL g, K(=p)-contiguous)
    {
        dim3 grid((75 + 3) / 4, CAPS / 32, IMGS);
        gemm_f32_wmma<EPI_SCALE_ROW><<<grid, 128, 0, stream>>>(
            heatU, NCOL, HW,        // A[c,p] = heatU[c*3136 + i*196 + p]
            g,     HW,  EMB * HW,   // B[p,e] = g[i*470400 + e*196 + p]
            xa1,   EMB, CAPS * EMB,
            Sinv,  CAPS,
            CAPS, EMB, HW);
    }
    // 9) xa = xa1 @ fc_w^T + fc_b   (M=1024, N=2400, K=2400)
    {
        dim3 grid((75 + 3) / 4, (IMGS * CAPS) / 32, 1);
        gemm_f32_wmma<EPI_BIAS><<<grid, 128, 0, stream>>>(
            xa1,  EMB, 0,
            fc_w, EMB, 0,
            out_xa, EMB, 0,
            fc_b, 0,
            IMGS * CAPS, EMB, EMB);
    }
    // 10) pass-through g output
    hipMemcpyAsync(out_g, (const void*)g,
                   (size_t)IMGS * EMB * HW * sizeof(float),
                   hipMemcpyDeviceToDevice, stream);
}